// FNOGNO_40570261078579
// MI455X (gfx1250) — compile-verified
//
#include <hip/hip_runtime.h>
#include <hip/hip_bf16.h>
#include <math.h>

typedef float v2f __attribute__((ext_vector_type(2)));
typedef float v8f __attribute__((ext_vector_type(8)));

#define GSZ   32
#define G3    32768
#define MODES 8
#define NLYR  4
#define HID   64
#define NOUT  16384
#define RAD2  (0.033f*0.033f)

__device__ __forceinline__ v8f v8zero() {
    v8f v;
#pragma unroll
    for (int i = 0; i < 8; ++i) v[i] = 0.0f;
    return v;
}

__device__ __forceinline__ v8f wmma4(v2f a, v2f b, v8f c) {
    // D = A(16x4 f32) * B(4x16 f32) + C(16x16 f32)
    return __builtin_amdgcn_wmma_f32_16x16x4_f32(false, a, false, b, (short)0, c, false, false);
}

__device__ __forceinline__ float gelu_f(float x) {
    return 0.5f * x * (1.0f + erff(x * 0.70710678118654752440f));
}

// ---------------------------------------------------------------------------
// Generic (complex) GEMM with WMMA f32 16x16x4.
// C[m,n] = sum_k A[m,k]*B[k,n]  (complex via separate Re/Im planes)
// Block: 128 threads = 4 waves. Block tile: 32 (M) x 64 (N).
// grid = (ceil(N/64), ceil(M/32), batch)
// ---------------------------------------------------------------------------
template<bool A_IM, bool B_IM, bool OUT_IM, bool BIAS, bool ADD, bool GELU>
__global__ __launch_bounds__(128)
void cgemm_kernel(const float* __restrict__ ARe, const float* __restrict__ AIm,
                  int lda, long long strA,
                  const float* __restrict__ BRe, const float* __restrict__ BIm,
                  int ldb, long long strB,
                  float* __restrict__ CRe, float* __restrict__ CIm,
                  int ldc, long long strC,
                  const float* __restrict__ bias, const float* __restrict__ addp,
                  int Mdim, int Ndim, int Kdim)
{
    const long long bz = blockIdx.z;
    ARe += bz * strA;
    BRe += bz * strB;
    CRe += bz * strC;
    const float* AImp = A_IM ? (AIm + bz * strA) : nullptr;
    const float* BImp = B_IM ? (BIm + bz * strB) : nullptr;
    float*       CImp = OUT_IM ? (CIm + bz * strC) : nullptr;
    const float* addq = ADD ? (addp + bz * strC) : nullptr;

    const int m0   = blockIdx.y * 32;
    const int wave = threadIdx.x >> 5;
    const int lane = threadIdx.x & 31;
    const int lm   = lane & 15;
    const int lh   = lane >> 4;
    const int n0   = blockIdx.x * 64 + wave * 16;
    const int row0 = m0 + lm;
    const int row1 = m0 + 16 + lm;
    const int col  = n0 + lm;

    v8f c0r = v8zero(), c1r = v8zero(), c0i = v8zero(), c1i = v8zero();

    for (int k0 = 0; k0 < Kdim; k0 += 4) {
        const int ka = k0 + 2 * lh;
        v2f a0r, a1r, a0i, a1i, br, bi;
#pragma unroll
        for (int j = 0; j < 2; ++j) {
            const int  k   = ka + j;
            const bool kin = (k < Kdim);
            const bool r0in = kin && (row0 < Mdim);
            const bool r1in = kin && (row1 < Mdim);
            const bool cin  = kin && (col < Ndim);
            a0r[j] = r0in ? ARe[(long long)row0 * lda + k] : 0.0f;
            a1r[j] = r1in ? ARe[(long long)row1 * lda + k] : 0.0f;
            br[j]  = cin  ? BRe[(long long)k * ldb + col]  : 0.0f;
            if (A_IM) {
                a0i[j] = r0in ? AImp[(long long)row0 * lda + k] : 0.0f;
                a1i[j] = r1in ? AImp[(long long)row1 * lda + k] : 0.0f;
            }
            if (B_IM) {
                bi[j] = cin ? BImp[(long long)k * ldb + col] : 0.0f;
            }
        }
        // Re += ARe*BRe (- AIm*BIm)
        c0r = wmma4(a0r, br, c0r);
        c1r = wmma4(a1r, br, c1r);
        if (A_IM && B_IM) {
            c0r = wmma4(-a0i, bi, c0r);
            c1r = wmma4(-a1i, bi, c1r);
        }
        if (OUT_IM) {
            if (A_IM) { c0i = wmma4(a0i, br, c0i); c1i = wmma4(a1i, br, c1i); }
            if (B_IM) { c0i = wmma4(a0r, bi, c0i); c1i = wmma4(a1r, bi, c1i); }
        }
    }

    if (col < Ndim) {
        const float bb = BIAS ? bias[col] : 0.0f;
#pragma unroll
        for (int v = 0; v < 8; ++v) {
            const int r0 = m0 + v + 8 * lh;
            const int r1 = r0 + 16;
            if (r0 < Mdim) {
                float x = c0r[v];
                if (BIAS) x += bb;
                if (ADD)  x += addq[(long long)r0 * ldc + col];
                if (GELU) x = gelu_f(x);
                CRe[(long long)r0 * ldc + col] = x;
                if (OUT_IM) CImp[(long long)r0 * ldc + col] = c0i[v];
            }
            if (r1 < Mdim) {
                float x = c1r[v];
                if (BIAS) x += bb;
                if (ADD)  x += addq[(long long)r1 * ldc + col];
                if (GELU) x = gelu_f(x);
                CRe[(long long)r1 * ldc + col] = x;
                if (OUT_IM) CImp[(long long)r1 * ldc + col] = c1i[v];
            }
        }
    }
}

// ---------------------------------------------------------------------------
// Init truncated DFT matrices (complex planes) into workspace.
// dft layout (floats): DzRe[8*32]@0, DzIm@256, DxyRe[16*32]@512, DxyIm@1024,
//                      ExyRe[32*16]@1536, ExyIm@2048, AzRe[32*8]@2560, AzIm@2816
// ---------------------------------------------------------------------------
__global__ void init_dft_kernel(float* __restrict__ dft)
{
    const int t = blockIdx.x * blockDim.x + threadIdx.x;
    const float TP = 6.283185307179586f;
    const float s = 1.0f / 32.0f;
    if (t < 256) {                       // Dz[k=8][z=32]: e^{-i}/32 (rfft axis)
        int k = t / 32, z = t % 32;
        int ph = (k * z) & 31;
        float th = TP * (float)ph / 32.0f;
        dft[t] = cosf(th) * s; dft[256 + t] = -sinf(th) * s;
    } else if (t < 768) {                // Dxy[m=16][x=32]: modes {0..7,24..31}, e^{-i}/32
        int u = t - 256; int m = u / 32, x = u % 32;
        int km = (m < 8) ? m : (m + 16);
        int ph = (km * x) & 31;
        float th = TP * (float)ph / 32.0f;
        dft[512 + u] = cosf(th) * s; dft[1024 + u] = -sinf(th) * s;
    } else if (t < 1280) {               // Exy[x=32][m=16]: e^{+i}, no scale
        int u = t - 768; int x = u / 16, m = u % 16;
        int km = (m < 8) ? m : (m + 16);
        int ph = (km * x) & 31;
        float th = TP * (float)ph / 32.0f;
        dft[1536 + u] = cosf(th); dft[2048 + u] = sinf(th);
    } else if (t < 1536) {               // Az[z=32][k=8]: w_k e^{+i}, w0=1 else 2 (c2r)
        int u = t - 1280; int z = u / 8, k = u % 8;
        float w = (k == 0) ? 1.0f : 2.0f;
        int ph = (k * z) & 31;
        float th = TP * (float)ph / 32.0f;
        dft[2560 + u] = w * cosf(th); dft[2816 + u] = w * sinf(th);
    }
}

// Pack lifting input [f(3), in_p(3)] -> [G^3, 6]
__global__ void pack_kernel(const float* __restrict__ f, const float* __restrict__ in_p,
                            float* __restrict__ apack)
{
    const int n = blockIdx.x * blockDim.x + threadIdx.x;
    if (n < G3) {
#pragma unroll
        for (int k = 0; k < 3; ++k) {
            apack[n * 6 + k]     = f[n * 3 + k];
            apack[n * 6 + 3 + k] = in_p[n * 3 + k];
        }
    }
}

// ---------------------------------------------------------------------------
// Spectral multiply: per mode position (kx,ky,kz) a 64x64 complex matvec.
// grid (16,16)=(ky,kx), block 256 = (kz=tid&7, og=tid>>3); thread does o=og,og+32.
// spec_w layout [l,c,i,o,mx,my,mz,2]; contiguous (mz,re/im) coalesces across kz lanes.
// ---------------------------------------------------------------------------
__global__ __launch_bounds__(256)
void spectral_kernel(const float* __restrict__ FxRe, const float* __restrict__ FxIm,
                     const float* __restrict__ spec_w, int layer,
                     float* __restrict__ SRe, float* __restrict__ SIm)
{
    __shared__ float sFr[512];
    __shared__ float sFi[512];
    const int ky = blockIdx.x, kx = blockIdx.y;
    const int corner = ((kx >= 8) ? 2 : 0) + ((ky >= 8) ? 1 : 0);
    const int mx = kx & 7, my = ky & 7;
    const long long posb = (long long)(kx * 16 + ky) * 512;

    for (int t = threadIdx.x; t < 512; t += 256) {
        sFr[t] = FxRe[posb + t];   // [kz][i]
        sFi[t] = FxIm[posb + t];
    }
    __syncthreads();

    const int kz = threadIdx.x & 7;
    const int og = threadIdx.x >> 3;          // 0..31
    const long long base = (long long)(layer * 4 + corner) * 4194304LL  // 64*64*512*2
                         + (long long)mx * 128 + (long long)my * 16 + (long long)kz * 2;
    float ar0 = 0.f, ai0 = 0.f, ar1 = 0.f, ai1 = 0.f;
    for (int i = 0; i < 64; ++i) {
        const float fr = sFr[kz * 64 + i];
        const float fi = sFi[kz * 64 + i];
        const long long ib = base + (long long)i * 65536;
        const long long i0 = ib + (long long)og * 1024;
        const long long i1 = i0 + 32LL * 1024;
        const float wr0 = spec_w[i0], wi0 = spec_w[i0 + 1];
        const float wr1 = spec_w[i1], wi1 = spec_w[i1 + 1];
        ar0 += fr * wr0 - fi * wi0;  ai0 += fr * wi0 + fi * wr0;
        ar1 += fr * wr1 - fi * wi1;  ai1 += fr * wi1 + fi * wr1;
    }
    SRe[posb + kz * 64 + og]      = ar0;
    SIm[posb + kz * 64 + og]      = ai0;
    SRe[posb + kz * 64 + og + 32] = ar1;
    SIm[posb + kz * 64 + og + 32] = ai1;
}

// ---------------------------------------------------------------------------
// Fused GNO: one block per output point; 27 edges (padded to 32 rows).
// MLP 6->512->256->64 with WMMA; hidden-512 chunked through LDS (4x128).
// Then gather latent + masked mean -> gno_out[p, 64].
// ---------------------------------------------------------------------------
__global__ __launch_bounds__(256)
void gno_kernel(const float* __restrict__ out_p, const float* __restrict__ latent,
                const float* __restrict__ w1, const float* __restrict__ b1,
                const float* __restrict__ w2, const float* __restrict__ b2,
                const float* __restrict__ w3, const float* __restrict__ b3,
                float* __restrict__ gno_out)
{
    __shared__ float sA[32 * 8];        //  1 KB  edge features (padded K=8)
    __shared__ float sH1[32 * 128];     // 16 KB  hidden1 chunk
    __shared__ float sH2[32 * 256];     // 32 KB  hidden2
    __shared__ float sK[32 * 64];       //  8 KB  kernel values
    __shared__ int   sFlat[32];
    __shared__ float sMask[32];

    const int p   = blockIdx.x;
    const int tid = threadIdx.x;
    const float px = out_p[p * 3 + 0];
    const float py = out_p[p * 3 + 1];
    const float pz = out_p[p * 3 + 2];

    if (tid < 256) sA[tid] = 0.0f;
    if (tid < 32) { sMask[tid] = 0.0f; sFlat[tid] = 0; }
    __syncthreads();

    if (tid < 27) {
        const int dx = tid / 9 - 1, dy = (tid / 3) % 3 - 1, dz = tid % 3 - 1;
        const int ix = (int)rintf(px * 31.0f) + dx;
        const int iy = (int)rintf(py * 31.0f) + dy;
        const int iz = (int)rintf(pz * 31.0f) + dz;
        const bool inb = (ix >= 0) & (ix < 32) & (iy >= 0) & (iy < 32) & (iz >= 0) & (iz < 32);
        const int cx = min(max(ix, 0), 31), cy = min(max(iy, 0), 31), cz = min(max(iz, 0), 31);
        const float hg = 1.0f / 31.0f;
        const float yx = cx * hg, yy = cy * hg, yz = cz * hg;
        const float ex = px - yx, ey = py - yy, ez = pz - yz;
        const float d2 = ex * ex + ey * ey + ez * ez;
        sFlat[tid] = (cx * 32 + cy) * 32 + cz;
        sMask[tid] = (inb && d2 <= RAD2) ? 1.0f : 0.0f;
        sA[tid * 8 + 0] = yx; sA[tid * 8 + 1] = yy; sA[tid * 8 + 2] = yz;
        sA[tid * 8 + 3] = px; sA[tid * 8 + 4] = py; sA[tid * 8 + 5] = pz;
    }
    __syncthreads();

    const int wave = tid >> 5, lane = tid & 31, lm = lane & 15, lh = lane >> 4;

    v8f g2[2][2];
    g2[0][0] = v8zero(); g2[0][1] = v8zero(); g2[1][0] = v8zero(); g2[1][1] = v8zero();

    for (int kc = 0; kc < 4; ++kc) {
        // ---- GEMM1 chunk: [32,8(K=6)] x [8,128] -> gelu -> sH1 ----
        {
            const int colg = kc * 128 + wave * 16 + lm;   // global hidden1 col
            v8f acc0 = v8zero(), acc1 = v8zero();
            for (int k0 = 0; k0 < 8; k0 += 4) {
                const int k = k0 + 2 * lh;
                v2f a0, a1, b;
                a0[0] = sA[lm * 8 + k];        a0[1] = sA[lm * 8 + k + 1];
                a1[0] = sA[(16 + lm) * 8 + k]; a1[1] = sA[(16 + lm) * 8 + k + 1];
                b[0] = (k     < 6) ? w1[k * 512 + colg]       : 0.0f;
                b[1] = (k + 1 < 6) ? w1[(k + 1) * 512 + colg] : 0.0f;
                acc0 = wmma4(a0, b, acc0);
                acc1 = wmma4(a1, b, acc1);
            }
            const int lcol = wave * 16 + lm;
            const float bb = b1[colg];
#pragma unroll
            for (int v = 0; v < 8; ++v) {
                const int r = v + 8 * lh;
                sH1[r * 128 + lcol]        = gelu_f(acc0[v] + bb);
                sH1[(16 + r) * 128 + lcol] = gelu_f(acc1[v] + bb);
            }
        }
        __syncthreads();
        // ---- GEMM2 partial: K-chunk 128 from sH1, B = w2 ----
        for (int k0 = 0; k0 < 128; k0 += 4) {
            const int k  = k0 + 2 * lh;
            const int kg = kc * 128 + k;
            v2f a0, a1;
            a0[0] = sH1[lm * 128 + k];        a0[1] = sH1[lm * 128 + k + 1];
            a1[0] = sH1[(16 + lm) * 128 + k]; a1[1] = sH1[(16 + lm) * 128 + k + 1];
#pragma unroll
            for (int nts = 0; nts < 2; ++nts) {
                const int colg = nts * 128 + wave * 16 + lm;
                v2f b;
                b[0] = w2[kg * 256 + colg];
                b[1] = w2[(kg + 1) * 256 + colg];
                g2[0][nts] = wmma4(a0, b, g2[0][nts]);
                g2[1][nts] = wmma4(a1, b, g2[1][nts]);
            }
        }
        __syncthreads();
    }
    // ---- GEMM2 epilogue: gelu(acc + b2) -> sH2 ----
#pragma unroll
    for (int nts = 0; nts < 2; ++nts) {
        const int colg = nts * 128 + wave * 16 + lm;
        const float bb = b2[colg];
#pragma unroll
        for (int v = 0; v < 8; ++v) {
            const int r = v + 8 * lh;
            sH2[r * 256 + colg]        = gelu_f(g2[0][nts][v] + bb);
            sH2[(16 + r) * 256 + colg] = gelu_f(g2[1][nts][v] + bb);
        }
    }
    __syncthreads();
    // ---- GEMM3: [32,256] x [256,64] -> sK ----
    {
        const int mt = wave >> 2, nt = wave & 3;
        const int colg = nt * 16 + lm;
        v8f acc = v8zero();
        for (int k0 = 0; k0 < 256; k0 += 4) {
            const int k = k0 + 2 * lh;
            v2f a, b;
            a[0] = sH2[(mt * 16 + lm) * 256 + k];
            a[1] = sH2[(mt * 16 + lm) * 256 + k + 1];
            b[0] = w3[k * 64 + colg];
            b[1] = w3[(k + 1) * 64 + colg];
            acc = wmma4(a, b, acc);
        }
        const float bb = b3[colg];
#pragma unroll
        for (int v = 0; v < 8; ++v)
            sK[(mt * 16 + v + 8 * lh) * 64 + colg] = acc[v] + bb;
    }
    __syncthreads();
    // ---- gather + masked mean ----
    if (tid < 64) {
        float num = 0.0f, den = 0.0f;
        for (int j = 0; j < 27; ++j) {
            const float m = sMask[j];
            den += m;
            num += m * sK[j * 64 + tid] * latent[(long long)sFlat[j] * 64 + tid];
        }
        gno_out[(long long)p * 64 + tid] = num / fmaxf(den, 1.0f);
    }
}

// Final projection GEMV: out[p] = b + proj_mid[p,:] . w
__global__ void proj2_kernel(const float* __restrict__ mid, const float* __restrict__ w,
                             const float* __restrict__ b, float* __restrict__ out, int n)
{
    const int p = blockIdx.x * blockDim.x + threadIdx.x;
    if (p < n) {
        float s = b[0];
        for (int c = 0; c < 256; ++c) s += mid[(long long)p * 256 + c] * w[c];
        out[p] = s;
    }
}

// ---------------------------------------------------------------------------
// Host launcher
// ---------------------------------------------------------------------------
extern "C" void kernel_launch(void* const* d_in, const int* in_sizes, int n_in,
                              void* d_out, int out_size, void* d_ws, size_t ws_size,
                              hipStream_t stream)
{
    const float* in_p    = (const float*)d_in[0];
    const float* out_p   = (const float*)d_in[1];
    const float* f       = (const float*)d_in[2];
    const float* lift_w1 = (const float*)d_in[3];
    const float* lift_b1 = (const float*)d_in[4];
    const float* lift_w2 = (const float*)d_in[5];
    const float* lift_b2 = (const float*)d_in[6];
    const float* spec_w  = (const float*)d_in[7];
    const float* skip_w  = (const float*)d_in[8];
    const float* skip_b  = (const float*)d_in[9];
    const float* gno_w1  = (const float*)d_in[10];
    const float* gno_b1  = (const float*)d_in[11];
    const float* gno_w2  = (const float*)d_in[12];
    const float* gno_b2  = (const float*)d_in[13];
    const float* gno_w3  = (const float*)d_in[14];
    const float* gno_b3  = (const float*)d_in[15];
    const float* proj_w1 = (const float*)d_in[16];
    const float* proj_b1 = (const float*)d_in[17];
    const float* proj_w2 = (const float*)d_in[18];
    const float* proj_b2 = (const float*)d_in[19];
    float* out = (float*)d_out;
    float* ws  = (float*)d_ws;

    // Workspace layout (float offsets)
    float* lift_mid = ws;                      // 32768*256
    float* apack    = ws + 8388608;            // 32768*6
    float* hA       = ws + 8585216;            // 32768*64
    float* hB       = ws + 10682368;           // 32768*64
    float* hs       = ws + 12779520;           // 32768*64
    float* FzRe     = ws + 14876672;           // 32*32*8*64
    float* FzIm     = ws + 15400960;
    float* FyRe     = ws + 15925248;           // 32*16*8*64
    float* FyIm     = ws + 16187392;
    float* FxRe     = ws + 16449536;           // 16*16*8*64
    float* FxIm     = ws + 16580608;
    float* SRe      = ws + 16711680;           // 16*16*8*64
    float* SIm      = ws + 16842752;
    float* GyRe     = ws + 16973824;           // 32*16*8*64
    float* GyIm     = ws + 17235968;
    float* HzRe     = ws + 17498112;           // 32*32*8*64
    float* HzIm     = ws + 18022400;
    float* gno_out  = ws + 18546688;           // 16384*64
    float* pmid     = ws + 19595264;           // 16384*256
    float* dft      = ws + 23789568;           // 3072
    float* DzRe  = dft,        * DzIm  = dft + 256;
    float* DxyRe = dft + 512,  * DxyIm = dft + 1024;
    float* ExyRe = dft + 1536, * ExyIm = dft + 2048;
    float* AzRe  = dft + 2560, * AzIm  = dft + 2816;

    const dim3 blk(128);

    // 0) DFT matrices + pack lifting input
    init_dft_kernel<<<dim3(6), dim3(256), 0, stream>>>(dft);
    pack_kernel<<<dim3(128), dim3(256), 0, stream>>>(f, in_p, apack);

    // 1) Lifting MLP: [32768,6]->256(gelu)->64
    cgemm_kernel<false,false,false,true,false,true><<<dim3(4,1024,1), blk, 0, stream>>>(
        apack, nullptr, 6, 0, lift_w1, nullptr, 256, 0,
        lift_mid, nullptr, 256, 0, lift_b1, nullptr, G3, 256, 6);
    cgemm_kernel<false,false,false,true,false,false><<<dim3(1,1024,1), blk, 0, stream>>>(
        lift_mid, nullptr, 256, 0, lift_w2, nullptr, 64, 0,
        hA, nullptr, 64, 0, lift_b2, nullptr, G3, 64, 256);

    // 2) FNO layers (truncated-mode DFT as GEMMs)
    float* cur = hA;
    float* nxt = hB;
    for (int l = 0; l < NLYR; ++l) {
        // (a) forward DFT along z (real -> complex, kz 0..7), batched per (x,y)
        cgemm_kernel<true,false,true,false,false,false><<<dim3(1,1,1024), blk, 0, stream>>>(
            DzRe, DzIm, 32, 0, cur, nullptr, 64, 2048,
            FzRe, FzIm, 64, 512, nullptr, nullptr, 8, 64, 32);
        // (b) forward DFT along y, batched per x
        cgemm_kernel<true,true,true,false,false,false><<<dim3(8,1,32), blk, 0, stream>>>(
            DxyRe, DxyIm, 32, 0, FzRe, FzIm, 512, 16384,
            FyRe, FyIm, 512, 8192, nullptr, nullptr, 16, 512, 32);
        // (c) forward DFT along x (single GEMM)
        cgemm_kernel<true,true,true,false,false,false><<<dim3(128,1,1), blk, 0, stream>>>(
            DxyRe, DxyIm, 32, 0, FyRe, FyIm, 8192, 0,
            FxRe, FxIm, 8192, 0, nullptr, nullptr, 16, 8192, 32);
        // (d) spectral multiply (per-mode 64x64 complex)
        spectral_kernel<<<dim3(16,16), dim3(256), 0, stream>>>(FxRe, FxIm, spec_w, l, SRe, SIm);
        // (e) inverse DFT along x
        cgemm_kernel<true,true,true,false,false,false><<<dim3(128,1,1), blk, 0, stream>>>(
            ExyRe, ExyIm, 16, 0, SRe, SIm, 8192, 0,
            GyRe, GyIm, 8192, 0, nullptr, nullptr, 32, 8192, 16);
        // (f) inverse DFT along y, batched per x
        cgemm_kernel<true,true,true,false,false,false><<<dim3(8,1,32), blk, 0, stream>>>(
            ExyRe, ExyIm, 16, 0, GyRe, GyIm, 512, 8192,
            HzRe, HzIm, 512, 16384, nullptr, nullptr, 32, 512, 16);
        // (g) inverse c2r along z (real output), batched per (x,y)
        cgemm_kernel<true,true,false,false,false,false><<<dim3(1,1,1024), blk, 0, stream>>>(
            AzRe, AzIm, 8, 0, HzRe, HzIm, 64, 512,
            hs, nullptr, 64, 2048, nullptr, nullptr, 32, 64, 8);
        // (h) skip GEMM + hs + bias (+gelu except last layer)
        if (l < NLYR - 1) {
            cgemm_kernel<false,false,false,true,true,true><<<dim3(1,1024,1), blk, 0, stream>>>(
                cur, nullptr, 64, 0, skip_w + l * 4096, nullptr, 64, 0,
                nxt, nullptr, 64, 0, skip_b + l * 64, hs, G3, 64, 64);
        } else {
            cgemm_kernel<false,false,false,true,true,false><<<dim3(1,1024,1), blk, 0, stream>>>(
                cur, nullptr, 64, 0, skip_w + l * 4096, nullptr, 64, 0,
                nxt, nullptr, 64, 0, skip_b + l * 64, hs, G3, 64, 64);
        }
        float* t = cur; cur = nxt; nxt = t;
    }
    // after 4 layers, latent is back in hA (== cur)

    // 3) Fused GNO (neighbor search + kernel MLP + gather + masked mean)
    gno_kernel<<<dim3(NOUT), dim3(256), 0, stream>>>(
        out_p, cur, gno_w1, gno_b1, gno_w2, gno_b2, gno_w3, gno_b3, gno_out);

    // 4) Projection: 64 -> 256 (gelu) -> 1
    cgemm_kernel<false,false,false,true,false,true><<<dim3(4,512,1), blk, 0, stream>>>(
        gno_out, nullptr, 64, 0, proj_w1, nullptr, 256, 0,
        pmid, nullptr, 256, 0, proj_b1, nullptr, NOUT, 256, 64);
    proj2_kernel<<<dim3(64), dim3(256), 0, stream>>>(pmid, proj_w2, proj_b2, out, NOUT);
}